// BoundedAttention_68831145886221
// MI455X (gfx1250) — compile-verified
//
#include <hip/hip_runtime.h>
#include <hip/hip_bf16.h>
#include <math.h>

// ---------------------------------------------------------------------------
// Decode attention, MI455X (gfx1250, wave32).
//   B=16, S=4, H=16, D=128, cache KV=4096, kv_len=4100, NPART=4 flash splits.
// Memory-bound: ~1.07GB of K/V streamed once -> ~46us floor @ 23.3TB/s.
// Dot products use V_WMMA_F32_16X16X4_F32 (fp32 in, fp32 accum).
// Software-pipelined: next K/V tile is loaded into VGPRs while the current
// tile (already in LDS) is computed; +prefetch 2 tiles ahead.
// ---------------------------------------------------------------------------

#define BQ      16      // batch
#define SQ      4       // query len
#define NH      16      // heads
#define HD      128     // head dim
#define KVC     4096    // cache len
#define KVTOT   4100
#define NPART   4
#define TILE    16
#define WAVES   4
#define RS      132     // padded LDS row stride in floats (bank-conflict free)
#define EPS     1e-5f

typedef __attribute__((ext_vector_type(2))) float v2f;
typedef __attribute__((ext_vector_type(4))) float v4f;
typedef __attribute__((ext_vector_type(8))) float v8f;

// ---------------------------------------------------------------------------
// Kernel 1: per-head RMSNorm + RoPE for q,k; v relayout. One wave per (b,s,h).
// Output layouts: wq/wk/wv = [b][h][s][128]; wq pre-scaled by 1/sqrt(D).
// ---------------------------------------------------------------------------
__global__ __launch_bounds__(32) void prep_kernel(
    const float* __restrict__ q, const float* __restrict__ k,
    const float* __restrict__ v, const float* __restrict__ fc,
    const float* __restrict__ fs, const float* __restrict__ qw,
    const float* __restrict__ kw,
    float* __restrict__ wq, float* __restrict__ wk, float* __restrict__ wv)
{
    int bid = blockIdx.x;          // b*64 + s*16 + h
    int b = bid >> 6;
    int s = (bid >> 4) & 3;
    int h = bid & 15;
    int t = threadIdx.x;           // 0..31, each lane owns 4 dims (2 rope pairs)
    int d0 = t * 4;

    size_t inoff = ((size_t)(b * SQ + s)) * (NH * HD) + h * HD + d0;
    v4f qx = *(const v4f*)(q + inoff);
    v4f kx = *(const v4f*)(k + inoff);
    v4f vx = *(const v4f*)(v + inoff);

    float sq = qx[0]*qx[0] + qx[1]*qx[1] + qx[2]*qx[2] + qx[3]*qx[3];
    float sk = kx[0]*kx[0] + kx[1]*kx[1] + kx[2]*kx[2] + kx[3]*kx[3];
#pragma unroll
    for (int o = 16; o >= 1; o >>= 1) {
        sq += __shfl_xor(sq, o);
        sk += __shfl_xor(sk, o);
    }
    float rq = rsqrtf(sq * (1.0f / HD) + EPS);
    float rk = rsqrtf(sk * (1.0f / HD) + EPS);

    v4f wqv = *(const v4f*)(qw + d0);
    v4f wkv = *(const v4f*)(kw + d0);
    float qy[4], ky[4];
#pragma unroll
    for (int i = 0; i < 4; i++) {
        qy[i] = qx[i] * rq * wqv[i];
        ky[i] = kx[i] * rk * wkv[i];
    }

    float c0 = fc[s * 64 + 2 * t],     sn0 = fs[s * 64 + 2 * t];
    float c1 = fc[s * 64 + 2 * t + 1], sn1 = fs[s * 64 + 2 * t + 1];
    float qo[4], ko[4];
    qo[0] = qy[0] * c0 - qy[1] * sn0;  qo[1] = qy[0] * sn0 + qy[1] * c0;
    qo[2] = qy[2] * c1 - qy[3] * sn1;  qo[3] = qy[2] * sn1 + qy[3] * c1;
    ko[0] = ky[0] * c0 - ky[1] * sn0;  ko[1] = ky[0] * sn0 + ky[1] * c0;
    ko[2] = ky[2] * c1 - ky[3] * sn1;  ko[3] = ky[2] * sn1 + ky[3] * c1;

    const float scale = 0.08838834764831845f;   // 1/sqrt(128)
    size_t ooff = ((size_t)((b * NH + h) * SQ + s)) * HD + d0;
#pragma unroll
    for (int i = 0; i < 4; i++) {
        wq[ooff + i] = qo[i] * scale;
        wk[ooff + i] = ko[i];
        wv[ooff + i] = vx[i];
    }
}

// ---------------------------------------------------------------------------
// Kernel 2: flash-decoding attention.
// Grid (B*H, NPART), 128 threads (4 wave32). Each wave streams 16-row K/V
// tiles into its private LDS slice (pipelined through VGPRs), does QK^T and
// PV via WMMA f32 16x16x4, online softmax per q-column. Partition partials
// (m, l, acc[128]) are merged across the 4 waves in LDS -> workspace.
// ---------------------------------------------------------------------------
__global__ __launch_bounds__(128) void attn_kernel(
    const float* __restrict__ cache_k, const float* __restrict__ cache_v,
    const float* __restrict__ wq, const float* __restrict__ wk,
    const float* __restrict__ wv, float* __restrict__ part)
{
    // Per wave: [0..2111]=K tile rows, [2112..4223]=V tile rows (stride RS).
    // After the tile loop the first 520 floats are reused as the merge overlay:
    //   [q*128+d] = acc, [512+2q]=m, [512+2q+1]=lsum.
    __shared__ float lds[WAVES][2 * TILE * RS];

    int bh = blockIdx.x;
    int b = bh >> 4, h = bh & 15;
    int p = blockIdx.y;
    int w  = threadIdx.x >> 5;
    int l  = threadIdx.x & 31;
    int lh = l & 15;
    bool hi = l >= 16;

    float* myK = lds[w];
    float* myV = lds[w] + TILE * RS;

    // --- Q fragments (B-matrix of score WMMA): wave-invariant, kept in VGPRs.
    // lane l supplies Q[q=l%16][4*ks + (hi?2:0) + {0,1}]; q>=4 columns are 0.
    v2f qf[32];
    {
        const float* qbase =
            wq + ((size_t)(b * NH + h) * SQ + lh) * HD + (hi ? 2 : 0);
#pragma unroll
        for (int ks = 0; ks < 32; ks++) {
            if (lh < SQ) {
                qf[ks][0] = qbase[4 * ks];
                qf[ks][1] = qbase[4 * ks + 1];
            } else {
                qf[ks][0] = 0.0f;
                qf[ks][1] = 0.0f;
            }
        }
    }

    v8f acc[8];                       // out^T accumulators, 8 d-chunks of 16
#pragma unroll
    for (int c = 0; c < 8; c++)
#pragma unroll
        for (int r = 0; r < 8; r++) acc[c][r] = 0.0f;

    float m = -3.0e38f, lsum = 0.0f;  // per q-column (dup across lane halves)

    int base0  = p * 1024;
    int ntiles = (p == NPART - 1) ? 65 : 64;       // last part covers 4 new kv
    int itmax  = (ntiles + WAVES - 1) / WAVES;
    int limit  = KVC + 1 + (lh < SQ ? lh : SQ - 1);  // causal: kv < limit
    int f4     = hi ? 16 : 0;          // lane halves split each 512B row

    // source-row pointers for a tile (rows beyond kv range clamp; masked later)
    auto tile_ptrs = [&](int t, const float*& ksrc, const float*& vsrc) {
        int kvr = base0 + t * TILE + lh;
        if (kvr < KVC) {
            size_t off = (((size_t)b * KVC + kvr) * NH + h) * HD;
            ksrc = cache_k + off;
            vsrc = cache_v + off;
        } else {
            int r2 = kvr - KVC;
            if (r2 > SQ - 1) r2 = SQ - 1;
            size_t off = ((size_t)(b * NH + h) * SQ + r2) * HD;
            ksrc = wk + off;
            vsrc = wv + off;
        }
    };

    v4f kreg[16], vreg[16];            // next tile, staged through VGPRs
    auto load_tile = [&](int t) {
        const float* ksrc;
        const float* vsrc;
        tile_ptrs(t, ksrc, vsrc);
#pragma unroll
        for (int j = 0; j < 16; j++) {
            kreg[j] = *(const v4f*)(ksrc + (f4 + j) * 4);
            vreg[j] = *(const v4f*)(vsrc + (f4 + j) * 4);
        }
    };
    auto store_tile = [&]() {
#pragma unroll
        for (int j = 0; j < 16; j++) {
            *(v4f*)(myK + lh * RS + (f4 + j) * 4) = kreg[j];
            *(v4f*)(myV + lh * RS + (f4 + j) * 4) = vreg[j];
        }
    };

    load_tile(w);                      // prologue: tile 0 of this wave

    for (int it = 0; it < itmax; it++) {
        int t = it * WAVES + w;
        bool active = t < ntiles;
        int kbase = base0 + t * TILE;

        if (active) store_tile();      // regs(tile t) -> LDS

        // issue global loads for tile t+WAVES; in flight during compute
        int tn = (it + 1) * WAVES + w;
        if (tn < ntiles) load_tile(tn);

        // prefetch two tiles ahead into GL2 (global_prefetch_b8)
        int tp = (it + 2) * WAVES + w;
        if (tp < ntiles) {
            const float* kpf;
            const float* vpf;
            tile_ptrs(tp, kpf, vpf);
            __builtin_prefetch(kpf + f4 * 4, 0, 0);
            __builtin_prefetch(kpf + f4 * 4 + 32, 0, 0);
            __builtin_prefetch(vpf + f4 * 4, 0, 0);
            __builtin_prefetch(vpf + f4 * 4 + 32, 0, 0);
        }

        __syncthreads();

        if (active) {
            // ---- scores^T (M=kv, N=q): 32 WMMAs, 2 interleaved accumulators
            v8f sc0, sc1;
#pragma unroll
            for (int r = 0; r < 8; r++) { sc0[r] = 0.0f; sc1[r] = 0.0f; }
#pragma unroll
            for (int ks = 0; ks < 32; ks += 2) {
                v2f a0 = *(const v2f*)(myK + lh * RS + 4 * ks + (hi ? 2 : 0));
                v2f a1 =
                    *(const v2f*)(myK + lh * RS + 4 * (ks + 1) + (hi ? 2 : 0));
                sc0 = __builtin_amdgcn_wmma_f32_16x16x4_f32(
                    false, a0, false, qf[ks], (short)0, sc0, false, false);
                sc1 = __builtin_amdgcn_wmma_f32_16x16x4_f32(
                    false, a1, false, qf[ks + 1], (short)0, sc1, false, false);
            }
            v8f sc;
#pragma unroll
            for (int r = 0; r < 8; r++) sc[r] = sc0[r] + sc1[r];

            // ---- mask + online softmax (per q-column = per lane)
            int kvofs = hi ? 8 : 0;
            float tm = -3.0e38f;
#pragma unroll
            for (int r = 0; r < 8; r++) {
                int kva = kbase + r + kvofs;
                float sv = (kva < limit) ? sc[r] : -1.0e30f;
                sc[r] = sv;
                tm = fmaxf(tm, sv);
            }
            tm = fmaxf(tm, __shfl_xor(tm, 16));
            float mnew  = fmaxf(m, tm);
            float alpha = __expf(m - mnew);
            m = mnew;
            lsum *= alpha;
#pragma unroll
            for (int c = 0; c < 8; c++)
#pragma unroll
                for (int r = 0; r < 8; r++) acc[c][r] *= alpha;

            float P[8];
            float ps = 0.0f;
#pragma unroll
            for (int r = 0; r < 8; r++) {
                P[r] = __expf(sc[r] - mnew);
                ps += P[r];
            }
            ps += __shfl_xor(ps, 16);
            lsum += ps;

            // ---- rebuild P as B-fragments (4 kv-slices) via half-swaps
            float t0 = __shfl_xor(P[2], 16), t1 = __shfl_xor(P[3], 16);
            float t2 = __shfl_xor(P[6], 16), t3 = __shfl_xor(P[7], 16);
            float t4 = __shfl_xor(P[0], 16), t5 = __shfl_xor(P[1], 16);
            float t6 = __shfl_xor(P[4], 16), t7 = __shfl_xor(P[5], 16);
            v2f bf[4];
            bf[0][0] = hi ? t0   : P[0];  bf[0][1] = hi ? t1   : P[1];
            bf[1][0] = hi ? t2   : P[4];  bf[1][1] = hi ? t3   : P[5];
            bf[2][0] = hi ? P[2] : t4;    bf[2][1] = hi ? P[3] : t5;
            bf[3][0] = hi ? P[6] : t6;    bf[3][1] = hi ? P[7] : t7;

            // ---- PV: out^T(chunk) += V^T-slice x P-slice, 32 WMMAs
            //      (8 independent chunk chains -> plenty of ILP)
#pragma unroll
            for (int c = 0; c < 8; c++) {
#pragma unroll
                for (int kvs = 0; kvs < 4; kvs++) {
                    int r0 = 4 * kvs + (hi ? 2 : 0);
                    const float* vp = myV + r0 * RS + c * 16 + lh;
                    v2f av;
                    av[0] = vp[0];
                    av[1] = vp[RS];
                    acc[c] = __builtin_amdgcn_wmma_f32_16x16x4_f32(
                        false, av, false, bf[kvs], (short)0, acc[c],
                        false, false);
                }
            }
        }
    }

    // ---- merge the 4 waves' partials in LDS (overlay own staging region)
    if (lh < SQ) {
#pragma unroll
        for (int c = 0; c < 8; c++)
#pragma unroll
            for (int r = 0; r < 8; r++)
                lds[w][lh * HD + c * 16 + r + (hi ? 8 : 0)] = acc[c][r];
        if (!hi) {
            lds[w][512 + 2 * lh]     = m;
            lds[w][512 + 2 * lh + 1] = lsum;
        }
    }
    __syncthreads();

    for (int idx = threadIdx.x; idx < SQ * HD; idx += 128) {
        int qq = idx >> 7, d = idx & 127;
        float M = -3.0e38f;
#pragma unroll
        for (int ww = 0; ww < WAVES; ww++)
            M = fmaxf(M, lds[ww][512 + 2 * qq]);
        float L = 0.0f, A = 0.0f;
#pragma unroll
        for (int ww = 0; ww < WAVES; ww++) {
            float e = __expf(lds[ww][512 + 2 * qq] - M);
            L += e * lds[ww][512 + 2 * qq + 1];
            A += e * lds[ww][qq * HD + d];
        }
        float* pp = part + ((size_t)(bh * NPART + p) * SQ + qq) * (HD + 4);
        if (d == 0) { pp[0] = M; pp[1] = L; }
        pp[2 + d] = A;
    }
}

// ---------------------------------------------------------------------------
// Kernel 3: combine the NPART partition partials, write final output
// out[b][q][h*128+d].
// ---------------------------------------------------------------------------
__global__ __launch_bounds__(128) void merge_kernel(
    const float* __restrict__ part, float* __restrict__ out)
{
    int bh = blockIdx.x;
    int b = bh >> 4, h = bh & 15;
    int d = threadIdx.x;
#pragma unroll
    for (int q = 0; q < SQ; q++) {
        float mv[NPART], lv[NPART];
        float M = -3.0e38f;
#pragma unroll
        for (int p = 0; p < NPART; p++) {
            const float* pp =
                part + ((size_t)(bh * NPART + p) * SQ + q) * (HD + 4);
            mv[p] = pp[0];
            lv[p] = pp[1];
            M = fmaxf(M, mv[p]);
        }
        float L = 0.0f, A = 0.0f;
#pragma unroll
        for (int p = 0; p < NPART; p++) {
            const float* pp =
                part + ((size_t)(bh * NPART + p) * SQ + q) * (HD + 4);
            float e = __expf(mv[p] - M);
            L += e * lv[p];
            A += e * pp[2 + d];
        }
        out[((size_t)(b * SQ + q)) * (NH * HD) + h * HD + d] = A / L;
    }
}

// ---------------------------------------------------------------------------
extern "C" void kernel_launch(void* const* d_in, const int* in_sizes, int n_in,
                              void* d_out, int out_size, void* d_ws,
                              size_t ws_size, hipStream_t stream)
{
    const float* q  = (const float*)d_in[0];
    const float* k  = (const float*)d_in[1];
    const float* v  = (const float*)d_in[2];
    const float* fc = (const float*)d_in[3];
    const float* fs = (const float*)d_in[4];
    const float* ck = (const float*)d_in[5];
    const float* cv = (const float*)d_in[6];
    const float* qw = (const float*)d_in[7];
    const float* kw = (const float*)d_in[8];
    float* out = (float*)d_out;

    const size_t headblk = (size_t)BQ * NH * SQ * HD;   // 131072 floats
    float* wq   = (float*)d_ws;
    float* wk   = wq + headblk;
    float* wv   = wk + headblk;
    float* part = wv + headblk;   // 256*NPART*4*(128+4) floats (~2.2 MB)

    prep_kernel<<<dim3(BQ * SQ * NH), dim3(32), 0, stream>>>(
        q, k, v, fc, fs, qw, kw, wq, wk, wv);
    attn_kernel<<<dim3(BQ * NH, NPART), dim3(128), 0, stream>>>(
        ck, cv, wq, wk, wv, part);
    merge_kernel<<<dim3(BQ * NH), dim3(128), 0, stream>>>(part, out);
}